// DeformablePSROIAlign_91250875171199
// MI455X (gfx1250) — compile-verified
//
#include <hip/hip_runtime.h>
#include <hip/hip_bf16.h>

typedef __attribute__((ext_vector_type(2))) float v2f;
typedef __attribute__((ext_vector_type(8))) float v8f;

#define POOLED 7
#define SAMPLE 4
#define FH 128
#define FW 128
#define FC 256
#define NROI 128
#define KTOT (POOLED * POOLED * FC)   // 12544
#define NCOL (POOLED * POOLED * 2)    // 98
#define LDP 112                       // 7 tiles of 16, padded N for partials
#define KSPLIT 8
#define KSEG (KTOT / KSPLIT)          // 1568 (divisible by 16)

// ---------------------------------------------------------------------------
// PS-ROI pooling (both passes). One block per (n, ph, pw) bin; 64 threads,
// each thread handles 4 consecutive channels via float4 (coalesced 1KB/corner).
// Sample geometry is wave-uniform (same for all lanes in the block).
// ---------------------------------------------------------------------------
__global__ __launch_bounds__(64)
void psroi_pool_kernel(const float* __restrict__ feat,   // [B,H,W,C] channels-last
                       const float* __restrict__ rois,   // [N,5]
                       const float* __restrict__ trans,  // [N,98] or unused
                       float* __restrict__ out,          // [N,P,P,C]
                       int hasTrans) {
    const int bin = blockIdx.x;              // n*49 + ph*7 + pw
    const int n  = bin / (POOLED * POOLED);
    const int r  = bin % (POOLED * POOLED);
    const int ph = r / POOLED;
    const int pw = r % POOLED;
    const int t  = threadIdx.x;              // channel group 4t..4t+3

    const float* roi = rois + n * 5;
    const int   b  = (int)roi[0];
    const float x1 = rintf(roi[1]) * 0.0625f - 0.5f;
    const float y1 = rintf(roi[2]) * 0.0625f - 0.5f;
    const float x2 = (rintf(roi[3]) + 1.0f) * 0.0625f - 0.5f;
    const float y2 = (rintf(roi[4]) + 1.0f) * 0.0625f - 0.5f;
    const float rw = fmaxf(x2 - x1, 0.1f);
    const float rh = fmaxf(y2 - y1, 0.1f);
    const float bw = rw * (1.0f / POOLED);
    const float bh = rh * (1.0f / POOLED);
    const float sbw = bw * 0.25f;
    const float sbh = bh * 0.25f;

    float tx = 0.0f, ty = 0.0f;
    if (hasTrans) {                          // part_idx[p] == p since PART==POOLED
        tx = trans[n * NCOL + 0 * 49 + ph * POOLED + pw];  // * TRANS_STD(=1)
        ty = trans[n * NCOL + 1 * 49 + ph * POOLED + pw];
    }
    const float wstart = pw * bw + x1 + tx * rw;
    const float hstart = ph * bh + y1 + ty * rh;

    const float* fb = feat + (size_t)b * FH * FW * FC;
    float4 sum = make_float4(0.f, 0.f, 0.f, 0.f);
    int cnt = 0;

    for (int sh = 0; sh < SAMPLE; ++sh) {
        for (int sw = 0; sw < SAMPLE; ++sw) {
            const float w = wstart + sw * sbw;
            const float h = hstart + sh * sbh;
            // uniform branch across block (geometry is per-bin)
            const bool valid = (w >= -0.5f) && (w <= FW - 0.5f) &&
                               (h >= -0.5f) && (h <= FH - 0.5f);
            if (!valid) continue;
            const float wc = fminf(fmaxf(w, 0.0f), (float)(FW - 1));
            const float hc = fminf(fmaxf(h, 0.0f), (float)(FH - 1));
            const int w0 = (int)floorf(wc);
            const int h0 = (int)floorf(hc);
            const int w1 = min(w0 + 1, FW - 1);
            const int h1 = min(h0 + 1, FH - 1);
            const float dw = wc - (float)w0;
            const float dh = hc - (float)h0;
            const float c00 = (1.0f - dh) * (1.0f - dw);
            const float c01 = (1.0f - dh) * dw;
            const float c10 = dh * (1.0f - dw);
            const float c11 = dh * dw;
            const float4 v00 = ((const float4*)(fb + ((size_t)h0 * FW + w0) * FC))[t];
            const float4 v01 = ((const float4*)(fb + ((size_t)h0 * FW + w1) * FC))[t];
            const float4 v10 = ((const float4*)(fb + ((size_t)h1 * FW + w0) * FC))[t];
            const float4 v11 = ((const float4*)(fb + ((size_t)h1 * FW + w1) * FC))[t];
            sum.x += v00.x * c00 + v01.x * c01 + v10.x * c10 + v11.x * c11;
            sum.y += v00.y * c00 + v01.y * c01 + v10.y * c10 + v11.y * c11;
            sum.z += v00.z * c00 + v01.z * c01 + v10.z * c10 + v11.z * c11;
            sum.w += v00.w * c00 + v01.w * c01 + v10.w * c10 + v11.w * c11;
            ++cnt;
        }
    }
    const float inv = (cnt > 0) ? (1.0f / (float)cnt) : 0.0f;
    float4 o = make_float4(sum.x * inv, sum.y * inv, sum.z * inv, sum.w * inv);
    ((float4*)(out + (size_t)bin * FC))[t] = o;
}

// ---------------------------------------------------------------------------
// FC via V_WMMA_F32_16X16X4_F32: offset = flat[128,12544] @ W[12544,98].
// Split-K x8 into partials [KSPLIT,128,112]; deterministic reduce adds bias.
// One wave per (Mtile, Ntile, Ksplit). Branch-free inner loop: B column is
// clamped to stay in-bounds — lanes with col>=98 only pollute D columns >=98,
// which land in the LDP=112 padding and are never read by the reducer. This
// keeps EXEC all-1s (required by WMMA) and lets loads clause/pipeline.
// ---------------------------------------------------------------------------
__global__ __launch_bounds__(32)
void fc_wmma_kernel(const float* __restrict__ flat,  // [128, KTOT]
                    const float* __restrict__ Wm,    // [KTOT, NCOL]
                    float* __restrict__ part) {      // [KSPLIT, 128, LDP]
    const int nt   = blockIdx.x;      // 0..6
    const int mt   = blockIdx.y;      // 0..7
    const int ks   = blockIdx.z;      // 0..7
    const int lane = threadIdx.x;     // 0..31
    const int m0 = mt * 16, n0 = nt * 16;
    const int half  = lane >> 4;      // 0: K=0,1   1: K=2,3
    const int khalf = half * 2;
    const int row  = m0 + (lane & 15);           // A-matrix row for this lane
    const int col  = n0 + (lane & 15);           // B/D column for this lane
    const int colc = min(col, NCOL - 1);         // clamped B load column
    const int k0 = ks * KSEG;

    const float* aptr = flat + (size_t)row * KTOT + k0 + khalf;
    const float* bptr = Wm + (size_t)(k0 + khalf) * NCOL + colc;
    v8f acc = {};

    for (int k = 0; k < KSEG; k += 16) {
        #pragma unroll
        for (int u = 0; u < 4; ++u) {
            const float2 av = *(const float2*)(aptr + 4 * u);
            v2f a; a.x = av.x; a.y = av.y;
            v2f bf;
            bf.x = bptr[(4 * u)     * NCOL];
            bf.y = bptr[(4 * u + 1) * NCOL];
            // (neg_a, A, neg_b, B, c_mod, C, reuse_a, reuse_b)
            acc = __builtin_amdgcn_wmma_f32_16x16x4_f32(
                false, a, false, bf, (short)0, acc, false, false);
        }
        aptr += 16;
        bptr += 16 * NCOL;
    }

    // D layout: VGPR i -> (M = m0 + 8*half + i, N = col)
    float* pbase = part + (size_t)ks * NROI * LDP;
    const int dm = m0 + half * 8;
    #pragma unroll
    for (int i = 0; i < 8; ++i) {
        pbase[(size_t)(dm + i) * LDP + col] = acc[i];
    }
}

// Deterministic split-K reduction + bias.
__global__ __launch_bounds__(256)
void fc_reduce_kernel(const float* __restrict__ part,  // [KSPLIT, 128, LDP]
                      const float* __restrict__ bias,  // [NCOL]
                      float* __restrict__ offs) {      // [128, NCOL]
    const int idx = blockIdx.x * blockDim.x + threadIdx.x;
    if (idx >= NROI * NCOL) return;
    const int n = idx / NCOL;
    const int j = idx % NCOL;
    float s = bias[j];
    #pragma unroll
    for (int ks = 0; ks < KSPLIT; ++ks)
        s += part[((size_t)ks * NROI + n) * LDP + j];
    offs[idx] = s;
}

// ---------------------------------------------------------------------------
// Launch: pool1 -> d_out (staging; pass2 never reads it), WMMA FC, reduce,
// pool2 overwrites d_out with the final [N,7,7,256].
// ---------------------------------------------------------------------------
extern "C" void kernel_launch(void* const* d_in, const int* in_sizes, int n_in,
                              void* d_out, int out_size, void* d_ws, size_t ws_size,
                              hipStream_t stream) {
    const float* feat = (const float*)d_in[0];
    const float* rois = (const float*)d_in[1];
    const float* fc_w = (const float*)d_in[2];
    const float* fc_b = (const float*)d_in[3];
    float* out  = (float*)d_out;

    float* part = (float*)d_ws;                       // KSPLIT*128*LDP floats
    float* offs = part + (size_t)KSPLIT * NROI * LDP; // 128*98 floats

    const int nbins = NROI * POOLED * POOLED;         // 6272 blocks

    // Pass 1: PS-ROI pool without offsets, staged into d_out.
    psroi_pool_kernel<<<nbins, 64, 0, stream>>>(feat, rois, nullptr, out, 0);

    // FC: full-precision WMMA GEMM, split-K, then deterministic reduce.
    dim3 gg(7, 8, KSPLIT);
    fc_wmma_kernel<<<gg, 32, 0, stream>>>(out, fc_w, part);
    fc_reduce_kernel<<<(NROI * NCOL + 255) / 256, 256, 0, stream>>>(part, fc_b, offs);

    // Pass 2: deformable PS-ROI pool with predicted offsets -> final d_out.
    psroi_pool_kernel<<<nbins, 64, 0, stream>>>(feat, rois, offs, out, 1);
}